// GaussianBlur_31009663877753
// MI455X (gfx1250) — compile-verified
//
#include <hip/hip_runtime.h>
#include <hip/hip_bf16.h>

// Gaussian blur 71x71, sigma 10, on [16,3,512,512] fp32, depthwise.
// Strategy: separable (kernel is rank-1 outer product; 1D factors = row/col
// sums). Each 1D 71-tap conv is a banded GEMM done with V_WMMA_F32_16X16X4_F32:
//   C(16x16) = sum over 22 chunks of A(16x4) x B(4x16)
// where B is a precomputed band matrix (B[j,n] = vec[j-n]) stored in the exact
// per-lane WMMA register order, and A is a tile of input rows staged via LDS.
// Each pass writes its result transposed (LDS transpose -> coalesced b128
// stores), so one kernel serves both passes and orientation is restored.

typedef float v2f __attribute__((ext_vector_type(2)));
typedef float v8f __attribute__((ext_vector_type(8)));

#define IMG_W 512
#define N_IMG 48              // 16 * 3
#define KS 71
#define KCHUNKS 22            // 22*4 = 88 >= 16 + 70
#define LDS_STRIDE 132        // 128-float window + pad; 132 % 64 = 4 -> conflict-free A gathers
#define TB_STRIDE 20          // transpose buffer stride (80B, 16B aligned rows)

// ---------------------------------------------------------------------------
// Setup: derive 1D factors from the 2D kernel, build band matrices in ws.
//   band layout: band[chunk*64 + lane*2 + v], value = vec[4*chunk + K - n],
//   K = v + 2*(lane/16), n = lane%16, zero outside [0,71).
//   ws[0..1407]   : band for pass 1 (kw, blur along W)
//   ws[1408..2815]: band for pass 2 (kh, blur along H)
// ---------------------------------------------------------------------------
__global__ void build_bands(const float* __restrict__ k2d, float* __restrict__ ws) {
    __shared__ float kh[KS];
    __shared__ float kw[KS];
    int t = threadIdx.x;
    if (t < KS) {
        float s = 0.f;
        for (int b = 0; b < KS; ++b) s += k2d[t * KS + b];
        kh[t] = s;                       // row sums -> factor along H
    } else if (t < 2 * KS) {
        int b = t - KS;
        float s = 0.f;
        for (int a = 0; a < KS; ++a) s += k2d[a * KS + b];
        kw[b] = s;                       // col sums -> factor along W
    }
    __syncthreads();
    for (int idx = t; idx < 2 * KCHUNKS * 64; idx += blockDim.x) {
        int which = idx / (KCHUNKS * 64);        // 0 -> kw (pass1), 1 -> kh (pass2)
        int r     = idx % (KCHUNKS * 64);
        int c     = r >> 6;                      // chunk
        int e     = r & 63;
        int lane  = e >> 1;
        int v     = e & 1;
        int K     = v + ((lane >> 4) << 1);
        int n     = lane & 15;
        int d     = 4 * c + K - n;
        float val = (d >= 0 && d < KS) ? (which ? kh[d] : kw[d]) : 0.0f;
        ws[idx] = val;
    }
}

// ---------------------------------------------------------------------------
// One blur pass: horizontal 71-tap conv along the contiguous dim of each
// 512x512 image, output written transposed. 4 waves / block, 1 tile / wave.
// ---------------------------------------------------------------------------
__global__ __launch_bounds__(128) void blur_pass(const float* __restrict__ in,
                                                 float* __restrict__ outT,
                                                 const float* __restrict__ band) {
    __shared__ float lds[4][16 * LDS_STRIDE];    // 33792 B
    const int wave = threadIdx.x >> 5;
    const int lane = threadIdx.x & 31;
    const int tile = blockIdx.x * 4 + wave;      // 0 .. 49151
    const int img  = tile >> 10;
    const int ty   = (tile >> 5) & 31;
    const int tx   = tile & 31;
    const int y0   = ty << 4;
    const int x0   = tx << 4;
    const float* src = in  + (size_t)img * (IMG_W * IMG_W);
    float*       dst = outT + (size_t)img * (IMG_W * IMG_W);
    float* sl = lds[wave];

    // --- preload band matrix fragments (register-resident across k-loop) ---
    v2f Bv[KCHUNKS];
    {
        const v2f* bp = (const v2f*)band;
#pragma unroll
        for (int c = 0; c < KCHUNKS; ++c) Bv[c] = bp[c * 32 + lane];
    }

    // --- stage input window [x0-36, x0+92) for 16 rows into LDS ---
    const int wbase = x0 - 36;                   // multiple of 4 -> 16B aligned
    if (wbase >= 0 && wbase + 128 <= IMG_W) {
#pragma unroll
        for (int i = 0; i < 16; ++i) {
            const float* row = src + (size_t)(y0 + i) * IMG_W + wbase;
            float4 v = ((const float4*)row)[lane];
            ((float4*)(sl + i * LDS_STRIDE))[lane] = v;
        }
    } else {
        for (int i = 0; i < 16; ++i) {
            const float* row = src + (size_t)(y0 + i) * IMG_W;
#pragma unroll
            for (int j = 0; j < 4; ++j) {
                int c = wbase + lane + j * 32;
                sl[i * LDS_STRIDE + lane + j * 32] =
                    (c >= 0 && c < IMG_W) ? row[c] : 0.0f;
            }
        }
    }
    __builtin_amdgcn_wave_barrier();             // per-wave LDS hand-off (DS in-order)

    // --- banded GEMM: C[m][n] = sum_j in[y0+m][x0-35+j] * band[j][n] ---
    v8f C = {};
    const int m   = lane & 15;
    const int hi2 = (lane >> 4) << 1;            // 0 or 2: A K-pair select
    const float* arow = sl + m * LDS_STRIDE + 1 + hi2;  // window idx = j + 1
#pragma unroll
    for (int c = 0; c < KCHUNKS; ++c) {
        v2f A;
        A.x = arow[c * 4 + 0];
        A.y = arow[c * 4 + 1];
        C = __builtin_amdgcn_wmma_f32_16x16x4_f32(
                false, A, false, Bv[c], (short)0, C, false, false);
    }
    __builtin_amdgcn_wave_barrier();

    // --- transpose 16x16 tile in LDS, store coalesced (b128) transposed ---
    {
        float* tb = sl;                          // reuse staging buffer
        const int n   = lane & 15;
        const int mhi = (lane >> 4) << 3;        // 0 or 8
#pragma unroll
        for (int r = 0; r < 8; ++r)
            tb[n * TB_STRIDE + mhi + r] = C[r];  // tb[n][m] = C[m][n]
        __builtin_amdgcn_wave_barrier();
        const int p  = lane >> 1;
        const int qb = (lane & 1) << 3;
        float4 o0 = *(const float4*)(tb + p * TB_STRIDE + qb);
        float4 o1 = *(const float4*)(tb + p * TB_STRIDE + qb + 4);
        float* orow = dst + (size_t)(x0 + p) * IMG_W + (y0 + qb);
        ((float4*)orow)[0] = o0;
        ((float4*)orow)[1] = o1;
    }
}

// ---------------------------------------------------------------------------
// Fallback: direct 71x71 conv (only used if workspace is too small).
// ---------------------------------------------------------------------------
__global__ void naive_conv(const float* __restrict__ x, const float* __restrict__ k2d,
                           float* __restrict__ out) {
    int idx = blockIdx.x * blockDim.x + threadIdx.x;
    int xx  = idx & 511;
    int yy  = (idx >> 9) & 511;
    int img = idx >> 18;
    const float* p = x + (size_t)img * (IMG_W * IMG_W);
    float acc = 0.f;
    for (int a = 0; a < KS; ++a) {
        int sy = yy + a - 35;
        if (sy < 0 || sy >= IMG_W) continue;
        const float* row = p + (size_t)sy * IMG_W;
        for (int b = 0; b < KS; ++b) {
            int sx = xx + b - 35;
            if (sx < 0 || sx >= IMG_W) continue;
            acc += row[sx] * k2d[a * KS + b];
        }
    }
    out[idx] = acc;
}

extern "C" void kernel_launch(void* const* d_in, const int* in_sizes, int n_in,
                              void* d_out, int out_size, void* d_ws, size_t ws_size,
                              hipStream_t stream) {
    const float* x   = (const float*)d_in[0];   // [16,3,512,512]
    const float* k2d = (const float*)d_in[1];   // [71,71]
    float* out = (float*)d_out;
    float* ws  = (float*)d_ws;

    const size_t interm_elems = (size_t)N_IMG * IMG_W * IMG_W;   // 12,582,912
    const size_t need = (4096 + interm_elems) * sizeof(float);

    if (ws_size >= need) {
        float* band_w = ws;                     // 1408 floats
        float* band_h = ws + KCHUNKS * 64;      // 1408 floats
        float* interm = ws + 4096;

        build_bands<<<1, 256, 0, stream>>>(k2d, ws);

        const int nTiles = N_IMG * 32 * 32;     // 49152
        dim3 grid(nTiles / 4);
        // pass 1: blur along W with kw, write transposed into interm
        blur_pass<<<grid, 128, 0, stream>>>(x, interm, band_w);
        // pass 2: blur along (now-contiguous) H with kh, write transposed -> d_out
        blur_pass<<<grid, 128, 0, stream>>>(interm, out, band_h);
    } else {
        const int total = (int)((size_t)N_IMG * IMG_W * IMG_W);
        naive_conv<<<(total + 255) / 256, 256, 0, stream>>>(x, k2d, out);
    }
}